// PTJPEG_19791209300052
// MI455X (gfx1250) — compile-verified
//
#include <hip/hip_runtime.h>
#include <math.h>

typedef float v2f __attribute__((ext_vector_type(2)));
typedef float v8f __attribute__((ext_vector_type(8)));

__constant__ float c_luma[64] = {
    16,11,10,16,24,40,51,61,  12,12,14,19,26,58,60,55,
    14,13,16,24,40,57,69,56,  14,17,22,29,51,87,80,62,
    18,22,37,56,68,109,103,77, 24,35,55,64,81,104,113,92,
    49,64,78,87,103,121,120,101, 72,92,95,98,112,100,103,99};
__constant__ float c_chroma[64] = {
    17,18,24,47,99,99,99,99,  18,21,26,66,99,99,99,99,
    24,26,56,99,99,99,99,99,  47,66,99,99,99,99,99,99,
    99,99,99,99,99,99,99,99,  99,99,99,99,99,99,99,99,
    99,99,99,99,99,99,99,99,  99,99,99,99,99,99,99,99};

__device__ __forceinline__ void wave_lds_sync() {
    asm volatile("s_wait_dscnt 0x0" ::: "memory");
}

// One 16x16x16 matmul via 4 chained v_wmma_f32_16x16x4_f32.
// A = Dbig (block-diag of 8x8 DCT basis D, from LDS table Dl) or Dbig^T.
// B = X^T where X is stored row-major [16][17] in LDS (Bsrc): B[k][n=m] = Bsrc[m*17+k].
// kbase = 2*(lane/16): which K sub-pair this lane supplies.
template <bool TRANSA>
__device__ __forceinline__ v8f chain(const float* Bsrc, const float* Dl,
                                     int m, int mi, int mb, int kbase) {
    v8f acc = {0, 0, 0, 0, 0, 0, 0, 0};
#pragma unroll
    for (int s = 0; s < 4; ++s) {
        const int k0 = 4 * s + kbase;        // even, so k0 and k0+1 share a block
        const int kb = k0 >> 3;
        const int kf = k0 & 7;
        float d0, d1;
        if (TRANSA) { d0 = Dl[kf * 8 + mi]; d1 = Dl[(kf + 1) * 8 + mi]; }  // Dbig^T[m][k]
        else        { d0 = Dl[mi * 8 + kf]; d1 = Dl[mi * 8 + kf + 1]; }    // Dbig[m][k]
        const bool on = (kb == mb);          // block-diagonal mask
        v2f a; a.x = on ? d0 : 0.0f; a.y = on ? d1 : 0.0f;
        v2f b; b.x = Bsrc[m * 17 + k0]; b.y = Bsrc[m * 17 + k0 + 1];
        acc = __builtin_amdgcn_wmma_f32_16x16x4_f32(
            /*neg_a=*/false, a, /*neg_b=*/false, b,
            /*c_mod=*/(short)0, acc, /*reuse_a=*/false, /*reuse_b=*/false);
    }
    return acc;
}

// Store C/D-layout v8f to LDS as row-major [16][17]
__device__ __forceinline__ void store_cd(float* T, v8f u, int m, int hh) {
#pragma unroll
    for (int v = 0; v < 8; ++v) T[(v + 8 * hh) * 17 + m] = u[v];
}

// X (row-major in T0)  ->  S = Dbig*X*Dbig^T  ->  round(S/q)*q  ->  R = Dbig^T*S*Dbig
// Uses T0/T1 as ping-pong transpose buffers; returns R in C/D layout.
__device__ __forceinline__ v8f jpeg_transform(float* T0, float* T1, const float* Dl,
                                              const float* qv, int m, int mi, int mb,
                                              int hh) {
    const int kbase = 2 * hh;
    v8f u = chain<false>(T0, Dl, m, mi, mb, kbase);   // U = Dbig * X^T
    store_cd(T1, u, m, hh);  wave_lds_sync();
    v8f s = chain<false>(T1, Dl, m, mi, mb, kbase);   // S = Dbig * U^T = Dbig X Dbig^T
#pragma unroll
    for (int v = 0; v < 8; ++v) s[v] = rintf(s[v] / qv[v]) * qv[v];
    store_cd(T0, s, m, hh);  wave_lds_sync();
    v8f w = chain<true>(T0, Dl, m, mi, mb, kbase);    // W = Dbig^T * S^T
    store_cd(T1, w, m, hh);  wave_lds_sync();
    return chain<true>(T1, Dl, m, mi, mb, kbase);     // R = Dbig^T * W^T
}

__global__ __launch_bounds__(256) void PTJPEG_kernel(const float* __restrict__ im,
                                                     const int* __restrict__ qp,
                                                     float* __restrict__ out) {
    // per-wave: rgb 3*[16][17]=816 | Dl 64 | T0 272 | T1 272  -> 1424 floats
    __shared__ float smem[8 * 1424];
    const int lane = threadIdx.x & 31;
    const int wave = threadIdx.x >> 5;
    float* rgb = smem + wave * 1424;
    float* Dl  = rgb + 816;
    float* T0  = Dl + 64;
    float* T1  = T0 + 272;

    const int tileId = blockIdx.x * 8 + wave;   // 0..32767
    const int b  = tileId >> 12;                // image (4096 tiles per image)
    const int t  = tileId & 4095;
    const int h0 = (t >> 6) << 4;
    const int w0 = (t & 63) << 4;

    // ---- build 8x8 DCT basis table in LDS (2 entries per lane) ----
#pragma unroll
    for (int it = 0; it < 2; ++it) {
        int e = it * 32 + lane;                 // 0..63
        int r = e >> 3, c = e & 7;
        float v = 0.5f * cosf((float)((2 * c + 1) * r) * 0.19634954084936207f);
        Dl[e] = (r == 0) ? 0.35355339059327373f : v;
    }

    // ---- load 16x16x3 RGB tile into LDS ----
    const float* imb = im + (size_t)b * 3u * 1024u * 1024u;
#pragma unroll
    for (int it = 0; it < 6; ++it) {
        int idx = it * 32 + lane;               // 0..191 float4s
        int ch = idx >> 6, rem = idx & 63, r = rem >> 2, c4 = rem & 3;
        const float4 v = *reinterpret_cast<const float4*>(
            imb + ((size_t)ch << 20) + (size_t)(h0 + r) * 1024u + w0 + (c4 << 2));
        float* d = rgb + ch * 272 + r * 17 + (c4 << 2);
        d[0] = v.x; d[1] = v.y; d[2] = v.z; d[3] = v.w;
    }
    wave_lds_sync();

    const int m  = lane & 15;
    const int hh = lane >> 4;
    const int mi = m & 7, mb = m >> 3;

    // ---- quality-scaled quant tables (statically-indexed register arrays) ----
    const int qual = *qp;
    const float scale = (qual < 50) ? (5000.0f / (float)qual)
                                    : (200.0f - 2.0f * (float)qual);
    float qlv[8], qcv[8];
#pragma unroll
    for (int v = 0; v < 8; ++v) {
        float l = floorf((c_luma[v * 8 + mi] * scale + 50.0f) * 0.01f);
        float c = floorf((c_chroma[v * 8 + mi] * scale + 50.0f) * 0.01f);
        qlv[v] = fminf(fmaxf(l, 1.0f), 255.0f);
        qcv[v] = fminf(fmaxf(c, 1.0f), 255.0f);
    }

    // ---- luma matrix staged row-major into T0: X[m][j] = Y(m, j) ----
#pragma unroll
    for (int j = 0; j < 16; ++j) {
        float R = rgb[m * 17 + j], G = rgb[272 + m * 17 + j], B = rgb[544 + m * 17 + j];
        T0[m * 17 + j] = 255.0f * (0.299f * R + 0.587f * G + 0.114f * B) - 128.0f;
    }
    wave_lds_sync();
    v8f ry = jpeg_transform(T0, T1, Dl, qlv, m, mi, mb, hh);

    // ---- chroma: 2x2 mean + dither + floor; Cb cols 0-7, Cr cols 8-15,
    //      rows 8-15 duplicate rows 0-7 (all four 8x8 blocks transform independently)
#pragma unroll
    for (int q = 0; q < 8; ++q) {
        float cbs = 0.0f, crs = 0.0f;
#pragma unroll
        for (int d = 0; d < 4; ++d) {
            int dy = d >> 1, dx = d & 1;
            int rr = 2 * mi + dy, cc = 2 * q + dx;
            float R = rgb[rr * 17 + cc], G = rgb[272 + rr * 17 + cc], B = rgb[544 + rr * 17 + cc];
            cbs += -0.168736f * R - 0.331264f * G + 0.5f * B;
            crs +=  0.5f * R - 0.418688f * G - 0.081312f * B;
        }
        float dith = (float)((q & 1) + 1) * 0.25f;
        T0[m * 17 + q]     = floorf(cbs * (255.0f * 0.25f) + dith);
        T0[m * 17 + 8 + q] = floorf(crs * (255.0f * 0.25f) + dith);
    }
    wave_lds_sync();
    v8f rc = jpeg_transform(T0, T1, Dl, qcv, m, mi, mb, hh);

    // ---- stage results (rgb buffer is dead), recombine, store ----
    float* FY = rgb;          // [16][17]
    float* FC = rgb + 272;    // [16][17]; rows 0-7 hold Cb(cols0-7)/Cr(cols8-15)
    store_cd(FY, ry, m, hh);
    store_cd(FC, rc, m, hh);
    wave_lds_sync();

    const size_t obase = (size_t)b * 3u * 1024u * 1024u;
#pragma unroll
    for (int it = 0; it < 8; ++it) {
        int p = it * 32 + lane;
        int i = p >> 4, j = p & 15;
        float yv = FY[i * 17 + j] + 128.0f;
        float cb = FC[(i >> 1) * 17 + (j >> 1)];
        float cr = FC[(i >> 1) * 17 + 8 + (j >> 1)];
        float R  = yv + 1.402f * cr;
        float G  = yv - 0.344136f * cb - 0.714136f * cr;
        float Bv = yv + 1.772f * cb;
        size_t o = (size_t)(h0 + i) * 1024u + w0 + j;
        const float inv255 = 1.0f / 255.0f;
        out[obase + o]              = fminf(fmaxf(R,  0.0f), 255.0f) * inv255;
        out[obase + (1u << 20) + o] = fminf(fmaxf(G,  0.0f), 255.0f) * inv255;
        out[obase + (2u << 20) + o] = fminf(fmaxf(Bv, 0.0f), 255.0f) * inv255;
    }
}

extern "C" void kernel_launch(void* const* d_in, const int* in_sizes, int n_in,
                              void* d_out, int out_size, void* d_ws, size_t ws_size,
                              hipStream_t stream) {
    const float* im = (const float*)d_in[0];
    const int* quality = (const int*)d_in[1];
    float* out = (float*)d_out;
    // 8 images * (64x64) 16x16 tiles = 32768 tiles; 8 waves (tiles) per 256-thread block
    dim3 grid(4096), block(256);
    PTJPEG_kernel<<<grid, block, 0, stream>>>(im, quality, out);
}